// SliceRenderer_75273596830034
// MI455X (gfx1250) — compile-verified
//
#include <hip/hip_runtime.h>
#include <hip/hip_bf16.h>

// SliceRenderer for MI455X (gfx1250, wave32).
// Strategy:
//   Kernel 1 (1 block, 512 thr): eff_w + bitonic sort (desc, stable) + emit
//     per-gaussian quadratic coefficients in sorted order into d_ws:
//       exponent(g; y,x) = cy2*y^2 + cy*y + cx2*x^2 + cx*x + c1
//     with ln(eff_w) folded into c1 (masked -> -1e30 so exp -> 0).
//   Kernel 2: per wave, 16 pixels x 512 sorted gaussians. The exponent matrix
//     (16 gaussians x 16 pixels) is one V_WMMA_F32_16X16X4_F32 (A = gaussian
//     coeffs on M, B = pixel features on K=4, C preloaded with c1 broadcast).
//     D layout gives each lane 8 *consecutive sorted* gaussians for its pixel,
//     so front-to-back compositing is an in-register scan; the two 8-gaussian
//     halves merge with one ds_swizzle SWAPX16 exchange (segmented composite:
//     S = Sf + Pf*Sb, P = Pf*Pb).

#define EPS 1e-8f
#define KGAUSS 512
#define WIMG 256

typedef __attribute__((ext_vector_type(2))) float v2f;
typedef __attribute__((ext_vector_type(8))) float v8f;

__device__ __forceinline__ float swapx16(float x) {
    // ds_swizzle_b32 group-of-32: xor=0x10, or=0, and=0x1f -> 0x401F
    return __int_as_float(__builtin_amdgcn_ds_swizzle(__float_as_int(x), 0x401F));
}

__global__ __launch_bounds__(KGAUSS) void slice_prep_sort(
    const float* __restrict__ pos,    // K x 3
    const float* __restrict__ scl,    // K x 3
    const float* __restrict__ opa,    // K
    const float* __restrict__ inten,  // K
    const float* __restrict__ zt,     // 1
    float* __restrict__ ws)           // 6 x 512 floats
{
    __shared__ unsigned long long skey[KGAUSS];
    const int k = threadIdx.x;
    const float ztgt = zt[0];

    // effective weight (z-slice)
    {
        float pz = pos[k * 3 + 2];
        float sz = scl[k * 3 + 2] + EPS;
        float zd = (ztgt - pz) / sz;
        float eff = 0.0f;
        if (fabsf(zd) < 3.0f) eff = opa[k] * __expf(-0.5f * zd * zd);
        // eff >= 0 so float bits are order-preserving; secondary key keeps
        // argsort(-eff) stability (smaller index first on ties).
        unsigned fb = __float_as_uint(eff);
        skey[k] = ((unsigned long long)fb << 32) | (unsigned)(511 - k);
    }
    __syncthreads();

    // bitonic sort, descending
    for (unsigned span = 2; span <= KGAUSS; span <<= 1) {
        for (unsigned j = span >> 1; j > 0; j >>= 1) {
            unsigned i = threadIdx.x;
            unsigned ixj = i ^ j;
            if (ixj > i) {
                unsigned long long av = skey[i], bv = skey[ixj];
                bool desc = ((i & span) == 0);
                if (desc ? (av < bv) : (av > bv)) { skey[i] = bv; skey[ixj] = av; }
            }
            __syncthreads();
        }
    }

    // emit sorted coefficients
    int src = 511 - (int)(skey[k] & 0xffffffffull);
    float px = pos[src * 3 + 0];
    float py = pos[src * 3 + 1];
    float pz = pos[src * 3 + 2];
    float sx = scl[src * 3 + 0] + EPS;
    float sy = scl[src * 3 + 1] + EPS;
    float sz = scl[src * 3 + 2] + EPS;
    float zd = (ztgt - pz) / sz;
    float eff = 0.0f;
    if (fabsf(zd) < 3.0f) eff = opa[src] * __expf(-0.5f * zd * zd);

    float i2x = -0.5f / (sx * sx);
    float i2y = -0.5f / (sy * sy);
    float c1v = (eff > 0.0f) ? (i2x * px * px + i2y * py * py + __logf(eff))
                             : -1e30f;

    ws[0 * KGAUSS + k] = i2x;                // cy2
    ws[1 * KGAUSS + k] = px / (sx * sx);     // cy
    ws[2 * KGAUSS + k] = i2y;                // cx2
    ws[3 * KGAUSS + k] = py / (sy * sy);     // cx
    ws[4 * KGAUSS + k] = c1v;                // c1 (with ln eff_w folded)
    ws[5 * KGAUSS + k] = inten[src];         // sorted intensity
}

__global__ __launch_bounds__(256) void slice_render(
    const float* __restrict__ ws, float* __restrict__ out)
{
    const float* cy2 = ws + 0 * KGAUSS;
    const float* cy  = ws + 1 * KGAUSS;
    const float* cx2 = ws + 2 * KGAUSS;
    const float* cx  = ws + 3 * KGAUSS;
    const float* c1  = ws + 4 * KGAUSS;
    const float* si  = ws + 5 * KGAUSS;

    const int lane = threadIdx.x & 31;
    const int wave = threadIdx.x >> 5;
    const int half = lane >> 4;       // 0: K=0,1 (y terms) / gaussians r..r ; 1: K=2,3 (x terms)
    const int l15  = lane & 15;

    // warm WGP$/L2 with the whole sorted coefficient block (12 KB)
    __builtin_prefetch(ws + threadIdx.x * 16, 0, 1);

    // 16-pixel strip per wave (strips are row-aligned since W=256)
    const int p0 = (blockIdx.x * 8 + wave) * 16;
    const int p  = p0 + l15;                 // this lane's pixel (N = lane&15)
    const float fy = (float)(p >> 8);        // row    (vs positions[:,0], scales[:,0])
    const float fx = (float)(p & (WIMG - 1)); // column (vs positions[:,1], scales[:,1])

    // B: 4x16 pixel-feature matrix, constant over the whole loop.
    // lanes 0-15: K=0 -> y^2, K=1 -> y ; lanes 16-31: K=2 -> x^2, K=3 -> x
    v2f bmat;
    bmat.x = half ? fx * fx : fy * fy;
    bmat.y = half ? fx : fy;

    float T = 1.0f;    // transmittance (identical across the 2 lanes sharing a pixel)
    float acc = 0.0f;  // rendered accumulator

    for (int n0 = 0; n0 < KGAUSS; n0 += 16) {
        // A: 16x4 sorted-gaussian coefficients (M = gaussian n0 + (lane&15))
        const int g = n0 + l15;
        v2f amat;
        amat.x = half ? cx2[g] : cy2[g];
        amat.y = half ? cx[g]  : cy[g];

        // C: broadcast c1 per gaussian row; D row r on this lane is gaussian
        // gbase + r (M = r + 8*half).
        const int gbase = n0 + (half << 3);
        v8f cmat;
#pragma unroll
        for (int r = 0; r < 8; ++r) cmat[r] = c1[gbase + r];

        // exponent[g, p] = A x B + C  (one f32 WMMA, full precision)
        v8f d = __builtin_amdgcn_wmma_f32_16x16x4_f32(
            false, amat, false, bmat, (short)0, cmat, false, false);

        // in-register front-to-back composite over this lane's 8 sorted gaussians
        float S = 0.0f, P = 1.0f;
#pragma unroll
        for (int r = 0; r < 8; ++r) {
            float alpha = fminf(__expf(d[r]), 0.99f);  // clip(gauss*eff_w, 0, 0.99)
            S += P * alpha * si[gbase + r];
            P *= (1.0f - alpha);
        }

        // merge halves: low lanes hold gaussians n0..n0+7 (front),
        // high lanes hold n0+8..n0+15 (back)
        float So = swapx16(S);
        float Po = swapx16(P);
        float Sf = half ? So : S;
        float Pf = half ? Po : P;
        float Sb = half ? S : So;
        float Pb = half ? P : Po;

        acc += T * (Sf + Pf * Sb);
        T *= Pf * Pb;
    }

    if (lane < 16) out[p] = acc;  // coalesced 16-wide store per wave
}

extern "C" void kernel_launch(void* const* d_in, const int* in_sizes, int n_in,
                              void* d_out, int out_size, void* d_ws, size_t ws_size,
                              hipStream_t stream) {
    const float* positions = (const float*)d_in[0];  // 512 x 3
    const float* scales    = (const float*)d_in[1];  // 512 x 3
    const float* opacity   = (const float*)d_in[2];  // 512
    const float* intensity = (const float*)d_in[3];  // 512
    const float* z_target  = (const float*)d_in[4];  // 1
    float* out = (float*)d_out;                      // 1 x 256 x 256
    float* ws  = (float*)d_ws;                       // 6 x 512 floats used

    slice_prep_sort<<<1, KGAUSS, 0, stream>>>(positions, scales, opacity,
                                              intensity, z_target, ws);
    // 65536 pixels / (8 waves * 16 pixels) = 512 blocks of 256 threads
    slice_render<<<512, 256, 0, stream>>>(ws, out);
}